// ImprovedPointDecoder_7894149890579
// MI455X (gfx1250) — compile-verified
//
#include <hip/hip_runtime.h>

typedef __attribute__((ext_vector_type(2))) float v2f;
typedef __attribute__((ext_vector_type(8))) float v8f;

#define TP      32          // points per block
#define THREADS 256         // 8 waves (wave32)
#define K1      448         // 64+128+256 gathered input channels
#define N1      512         // hidden width
#define K2      512
#define N2      128         // output channels
#define XS      456         // X tile row stride (floats), even, !=0 mod 64
#define HS      520         // h tile row stride (floats)

// LDS layout (floats). X tile (32x456=14592) and h tile (32x520=16640) are
// overlaid in region [0,16640), separated by barriers.
#define BASE_OFF  (TP * HS)            // 16640 : int voxel bases [TP][3]
#define RED_OFF   (BASE_OFF + 96)      // LN partials: sum[32][8], sumsq[32][8]
#define STAT_OFF  (RED_OFF + 512)      // mean/rstd [32][2]
#define SMEM_FLTS (STAT_OFF + 64)      // 17312 floats = 69248 bytes

// ---------------- prep kernels (tiny, run each launch) ----------------

// WcT[j*448 + k] = sum_o w_s[c,o] * fw1[128*s + o, j]   (k -> (s,c))
__global__ __launch_bounds__(256) void build_wct(
    const float* __restrict__ w0, const float* __restrict__ w1,
    const float* __restrict__ w2, const float* __restrict__ fw1,
    float* __restrict__ WcT) {
  int t = blockIdx.x * 256 + threadIdx.x;
  if (t >= 448 * 512) return;
  int k = t % 448, j = t / 448;
  const float* wsrc; int s, c;
  if (k < 64)       { s = 0; c = k;       wsrc = w0; }
  else if (k < 192) { s = 1; c = k - 64;  wsrc = w1; }
  else              { s = 2; c = k - 192; wsrc = w2; }
  const float* wr = wsrc + c * 128;
  const float* fr = fw1 + (s * 128) * 512 + j;
  float acc = 0.f;
  for (int o = 0; o < 128; ++o) acc += wr[o] * fr[o * 512];
  WcT[t] = acc;   // t == j*448 + k
}

// fbc[j] = fb1[j] + sum_s sum_o b_s[o] * fw1[128*s + o, j]
__global__ __launch_bounds__(256) void build_fbc(
    const float* __restrict__ b0, const float* __restrict__ b1,
    const float* __restrict__ b2, const float* __restrict__ fw1,
    const float* __restrict__ fb1, float* __restrict__ fbc) {
  int j = blockIdx.x * 256 + threadIdx.x;
  if (j >= 512) return;
  float acc = fb1[j];
  for (int o = 0; o < 128; ++o) acc += b0[o] * fw1[o * 512 + j];
  for (int o = 0; o < 128; ++o) acc += b1[o] * fw1[(128 + o) * 512 + j];
  for (int o = 0; o < 128; ++o) acc += b2[o] * fw1[(256 + o) * 512 + j];
  fbc[j] = acc;
}

// fw2T[col*512 + k] = fw2[k*128 + col]
__global__ __launch_bounds__(256) void transpose_fw2(
    const float* __restrict__ fw2, float* __restrict__ fw2T) {
  int t = blockIdx.x * 256 + threadIdx.x;
  if (t >= 512 * 128) return;
  int col = t / 512, k = t % 512;
  fw2T[t] = fw2[k * 128 + col];
}

// ---------------- fused main kernel ----------------

__global__ __launch_bounds__(THREADS) void fused_point_decoder(
    const float* __restrict__ f0, const float* __restrict__ f1,
    const float* __restrict__ f2, const float* __restrict__ coords,
    const float* __restrict__ WcT, const float* __restrict__ fbc,
    const float* __restrict__ fw2T, const float* __restrict__ lng,
    const float* __restrict__ lnb, const float* __restrict__ fb2,
    float* __restrict__ out, int Npts, int total) {
  extern __shared__ float smem[];
  float* Xg   = smem;                          // [TP][XS] (gather + GEMM1)
  float* Hs   = smem;                          // [TP][HS] (h, overlaid)
  int*   bases = (int*)(smem + BASE_OFF);      // [TP][3]
  float* red  = smem + RED_OFF;                // [32][8] sum / +256 sumsq
  float* stat = smem + STAT_OFF;               // [32][2] mean,rstd

  const int tid = threadIdx.x;
  const int g0  = blockIdx.x * TP;

  // ---- phase 0: per-point voxel base offsets (nearest-voxel, like ref) ----
  if (tid < TP) {
    int g = g0 + tid;
    if (g >= total) g = total - 1;
    int b = g / Npts;
    float cx = coords[g * 3 + 0];
    float cy = coords[g * 3 + 1];
    float cz = coords[g * 3 + 2];
    {  // scale 0: C=64, D=H=W=64
      int ix = (int)fminf(fmaxf(cx * 64.f, 0.f), 63.f);
      int iy = (int)fminf(fmaxf(cy * 64.f, 0.f), 63.f);
      int iz = (int)fminf(fmaxf(cz * 64.f, 0.f), 63.f);
      bases[tid * 3 + 0] = b * (64 * 262144) + ix * 4096 + iy * 64 + iz;
    }
    {  // scale 1: C=128, D=H=W=32
      int ix = (int)fminf(fmaxf(cx * 32.f, 0.f), 31.f);
      int iy = (int)fminf(fmaxf(cy * 32.f, 0.f), 31.f);
      int iz = (int)fminf(fmaxf(cz * 32.f, 0.f), 31.f);
      bases[tid * 3 + 1] = b * (128 * 32768) + ix * 1024 + iy * 32 + iz;
    }
    {  // scale 2: C=256, D=H=W=16
      int ix = (int)fminf(fmaxf(cx * 16.f, 0.f), 15.f);
      int iy = (int)fminf(fmaxf(cy * 16.f, 0.f), 15.f);
      int iz = (int)fminf(fmaxf(cz * 16.f, 0.f), 15.f);
      bases[tid * 3 + 2] = b * (256 * 4096) + ix * 256 + iy * 16 + iz;
    }
  }
  __syncthreads();

  // ---- phase 1: gather 448 channels per point into LDS (L2-resident reads) ----
  for (int i = tid; i < TP * K1; i += THREADS) {
    int k = i >> 5, p = i & 31;
    float v;
    if (k < 64)       v = f0[bases[p * 3 + 0] + k * 262144];
    else if (k < 192) v = f1[bases[p * 3 + 1] + (k - 64) * 32768];
    else              v = f2[bases[p * 3 + 2] + (k - 192) * 4096];
    Xg[p * XS + k] = v;
  }
  __syncthreads();

  // ---- phase 2: GEMM1  h = relu(X @ Wc + fbc)  via v_wmma_f32_16x16x4_f32 ----
  const int w      = tid >> 5;           // wave 0..7 -> 64-wide column group
  const int lane   = tid & 31;
  const int lhalf  = lane & 15;
  const int koff   = (lane >> 4) << 1;   // 0 or 2 (A/B fragment K sub-offset)
  const int rowadd = (lane >> 4) << 3;   // 0 or 8 (C fragment M offset)
  const int colbase = w * 64;

  v8f acc[2][4];
  #pragma unroll
  for (int rt = 0; rt < 2; ++rt)
    #pragma unroll
    for (int ct = 0; ct < 4; ++ct) acc[rt][ct] = (v8f){};

  for (int k0 = 0; k0 < K1; k0 += 4) {
    v2f a[2];
    #pragma unroll
    for (int rt = 0; rt < 2; ++rt) {
      const float* ap = &Xg[(rt * 16 + lhalf) * XS + k0 + koff];
      a[rt][0] = ap[0]; a[rt][1] = ap[1];           // ds_load_b64
    }
    #pragma unroll
    for (int ct = 0; ct < 4; ++ct) {
      const float* bp = &WcT[(colbase + ct * 16 + lhalf) * 448 + k0 + koff];
      v2f bv; bv[0] = bp[0]; bv[1] = bp[1];         // global_load_b64
      #pragma unroll
      for (int rt = 0; rt < 2; ++rt)
        acc[rt][ct] = __builtin_amdgcn_wmma_f32_16x16x4_f32(
            false, a[rt], false, bv, (short)0, acc[rt][ct], false, false);
    }
  }
  __syncthreads();  // all X reads done; reuse region for h

  #pragma unroll
  for (int ct = 0; ct < 4; ++ct) {
    int col = colbase + ct * 16 + lhalf;
    float bb = fbc[col];
    #pragma unroll
    for (int rt = 0; rt < 2; ++rt)
      #pragma unroll
      for (int i = 0; i < 8; ++i) {
        float v = acc[rt][ct][i] + bb;
        Hs[(rt * 16 + i + rowadd) * HS + col] = v > 0.f ? v : 0.f;
      }
  }
  __syncthreads();

  // ---- phase 3: LayerNorm over 512 per point ----
  {
    int p = tid >> 3, sub = tid & 7;
    const float* hp = &Hs[p * HS + sub * 64];
    float s = 0.f, sq = 0.f;
    for (int c = 0; c < 64; ++c) { float v = hp[c]; s += v; sq += v * v; }
    red[p * 8 + sub] = s;
    red[256 + p * 8 + sub] = sq;
  }
  __syncthreads();
  if (tid < TP) {
    float s = 0.f, sq = 0.f;
    for (int j = 0; j < 8; ++j) { s += red[tid * 8 + j]; sq += red[256 + tid * 8 + j]; }
    float mean = s * (1.f / 512.f);
    float var  = sq * (1.f / 512.f) - mean * mean;
    stat[tid * 2 + 0] = mean;
    stat[tid * 2 + 1] = rsqrtf(var + 1e-5f);
  }
  __syncthreads();
  for (int i = tid; i < TP * N1; i += THREADS) {
    int p = i >> 9, c = i & 511;
    float v = Hs[p * HS + c];
    Hs[p * HS + c] = (v - stat[p * 2]) * stat[p * 2 + 1] * lng[c] + lnb[c];
  }
  __syncthreads();

  // ---- phase 4: GEMM2  out = h @ fw2 + fb2 ----
  v8f acc2[2];
  acc2[0] = (v8f){}; acc2[1] = (v8f){};
  const int rt2  = w & 1;
  const int ct2b = (w >> 1) << 1;          // {0,2,4,6}

  for (int k0 = 0; k0 < K2; k0 += 4) {
    v2f a;
    const float* ap = &Hs[(rt2 * 16 + lhalf) * HS + k0 + koff];
    a[0] = ap[0]; a[1] = ap[1];
    #pragma unroll
    for (int j = 0; j < 2; ++j) {
      const float* bp = &fw2T[((ct2b + j) * 16 + lhalf) * 512 + k0 + koff];
      v2f bv; bv[0] = bp[0]; bv[1] = bp[1];
      acc2[j] = __builtin_amdgcn_wmma_f32_16x16x4_f32(
          false, a, false, bv, (short)0, acc2[j], false, false);
    }
  }

  #pragma unroll
  for (int j = 0; j < 2; ++j) {
    int col = (ct2b + j) * 16 + lhalf;
    float bb = fb2[col];
    #pragma unroll
    for (int i = 0; i < 8; ++i) {
      int g = g0 + rt2 * 16 + i + rowadd;
      if (g < total) out[g * N2 + col] = acc2[j][i] + bb;
    }
  }
}

// ---------------- launcher ----------------

extern "C" void kernel_launch(void* const* d_in, const int* in_sizes, int n_in,
                              void* d_out, int out_size, void* d_ws, size_t ws_size,
                              hipStream_t stream) {
  const float* f0     = (const float*)d_in[0];
  const float* f1     = (const float*)d_in[1];
  const float* f2     = (const float*)d_in[2];
  const float* coords = (const float*)d_in[3];
  const float* w0     = (const float*)d_in[4];
  const float* b0     = (const float*)d_in[5];
  const float* w1     = (const float*)d_in[6];
  const float* b1     = (const float*)d_in[7];
  const float* w2     = (const float*)d_in[8];
  const float* b2     = (const float*)d_in[9];
  const float* fw1    = (const float*)d_in[10];
  const float* fb1    = (const float*)d_in[11];
  const float* lng    = (const float*)d_in[12];
  const float* lnb    = (const float*)d_in[13];
  const float* fw2    = (const float*)d_in[14];
  const float* fb2    = (const float*)d_in[15];

  float* ws   = (float*)d_ws;
  float* WcT  = ws;                       // 448*512 floats
  float* fbc  = WcT + 448 * 512;          // 512 floats
  float* fw2T = fbc + 512;                // 512*128 floats (~1.18 MB total)

  int B     = in_sizes[0] / (64 * 262144);         // elems of f0 / per-batch
  if (B <= 0) B = 1;
  int Npts  = in_sizes[3] / (3 * B);
  int total = B * Npts;

  build_wct<<<(448 * 512 + 255) / 256, 256, 0, stream>>>(w0, w1, w2, fw1, WcT);
  build_fbc<<<2, 256, 0, stream>>>(b0, b1, b2, fw1, fb1, fbc);
  transpose_fw2<<<(512 * 128 + 255) / 256, 256, 0, stream>>>(fw2, fw2T);

  int blocks = (total + TP - 1) / TP;
  size_t shmem = SMEM_FLTS * sizeof(float);
  fused_point_decoder<<<blocks, THREADS, shmem, stream>>>(
      f0, f1, f2, coords, WcT, fbc, fw2T, lng, lnb, fb2,
      (float*)d_out, Npts, total);
}